// Graphsage_88888643158466
// MI455X (gfx1250) — compile-verified
//
#include <hip/hip_runtime.h>

// ---------------- problem constants (from reference) ----------------
constexpr int N0 = 400000, N1 = 100000, N2 = 25000;
constexpr int E0 = 800000, E1 = 200000;
constexpr int DIN = 128, HID = 256, DOUT = 128;

typedef __attribute__((ext_vector_type(2))) float v2f;
typedef __attribute__((ext_vector_type(8))) float v8f;

// ---------------- zero fill ----------------
__global__ void zero_f32(float* __restrict__ p, long n) {
    long i = (long)blockIdx.x * blockDim.x + threadIdx.x;
    if (i < n) p[i] = 0.0f;
}

// ---------------- feat = concat(emb0[x0], emb1[x1], emb2[x2], emb3[x3]) ----------------
// one wave per row; lane handles 4 contiguous floats of the 128-wide row
__global__ void build_feat(const int* __restrict__ x,
                           const float* __restrict__ e0, const float* __restrict__ e1,
                           const float* __restrict__ e2, const float* __restrict__ e3,
                           float* __restrict__ feat) {
    int t = blockIdx.x * blockDim.x + threadIdx.x;
    int row = t >> 5, lane = t & 31;
    if (row >= N0) return;
    const float4* src;
    if (lane < 8)       src = (const float4*)(e0 + (long)x[row * 4 + 0] * 32) + lane;
    else if (lane < 12) src = (const float4*)(e1 + (long)x[row * 4 + 1] * 16) + (lane - 8);
    else if (lane < 16) src = (const float4*)(e2 + (long)x[row * 4 + 2] * 16) + (lane - 12);
    else                src = (const float4*)(e3 + (long)x[row * 4 + 3] * 64) + (lane - 16);
    ((float4*)(feat + (long)row * 128))[lane] = *src;
}

// ---------------- edge scatter: agg[dst] += h[src]; deg[dst] += 1 ----------------
// one wave per edge; WIDTH = 128 or 256 floats per row
template <int WIDTH>
__global__ void scatter_edges(const float* __restrict__ h, const int* __restrict__ srcA,
                              const int* __restrict__ dstA, float* __restrict__ agg,
                              float* __restrict__ deg, int nE) {
    int t = blockIdx.x * blockDim.x + threadIdx.x;
    int e = t >> 5, lane = t & 31;
    if (e >= nE) return;
    int s = srcA[e], d = dstA[e];
    const float* hp = h + (long)s * WIDTH;
    float* ap = agg + (long)d * WIDTH;
#pragma unroll
    for (int j = 0; j < WIDTH / 128; ++j) {
        float4 v = ((const float4*)hp)[lane + 32 * j];
        atomicAdd(ap + 4 * lane + 128 * j + 0, v.x);
        atomicAdd(ap + 4 * lane + 128 * j + 1, v.y);
        atomicAdd(ap + 4 * lane + 128 * j + 2, v.z);
        atomicAdd(ap + 4 * lane + 128 * j + 3, v.w);
    }
    if (lane == 0) atomicAdd(deg + d, 1.0f);
}

// ---------------- fused GEMM1: h1 = relu((feat+agg/max(deg,1)) @ W + (1+[deg>0])*b) ----------------
// block = 256 thr (8 waves), one 16-row M tile per block; wave w does N tiles w and w+8
__global__ void __launch_bounds__(256)
gemm1_fused(const float* __restrict__ feat, const float* __restrict__ agg0,
            const float* __restrict__ deg0, const float* __restrict__ W,
            const float* __restrict__ bias, float* __restrict__ h1) {
    __shared__ float As[16 * 132];  // stride 132 to avoid bank conflicts
    __shared__ float srow[16];
    const int m0 = blockIdx.x * 16;
    const int tid = threadIdx.x;
    // prologue: A_row = feat[row] + agg0[row]/max(deg,1)
#pragma unroll
    for (int i = 0; i < 8; ++i) {
        int idx = tid + i * 256;           // 16*128 elements
        int r = idx >> 7, c = idx & 127;
        long row = m0 + r;
        float inv = 1.0f / fmaxf(deg0[row], 1.0f);
        As[r * 132 + c] = feat[row * 128 + c] + agg0[row * 128 + c] * inv;
    }
    if (tid < 16) srow[tid] = (deg0[m0 + tid] > 0.0f) ? 2.0f : 1.0f;
    __syncthreads();

    const int lane = tid & 31, wave = tid >> 5;
    const int lo = lane & 15, hi = lane >> 4;
    const int nA = wave * 16, nB = (wave + 8) * 16;
    v8f c0 = {0, 0, 0, 0, 0, 0, 0, 0};
    v8f c1 = {0, 0, 0, 0, 0, 0, 0, 0};
    for (int k = 0; k < DIN; k += 4) {
        const int kk = k + 2 * hi;
        v2f a;
        a[0] = As[lo * 132 + kk];
        a[1] = As[lo * 132 + kk + 1];
        v2f b0, b1;
        b0[0] = W[kk * HID + nA + lo];
        b0[1] = W[(kk + 1) * HID + nA + lo];
        b1[0] = W[kk * HID + nB + lo];
        b1[1] = W[(kk + 1) * HID + nB + lo];
        c0 = __builtin_amdgcn_wmma_f32_16x16x4_f32(false, a, false, b0, (short)0, c0, false, false);
        c1 = __builtin_amdgcn_wmma_f32_16x16x4_f32(false, a, false, b1, (short)0, c1, false, false);
    }
#pragma unroll
    for (int r = 0; r < 8; ++r) {
        int rr = r + 8 * hi;
        long row = m0 + rr;
        float s = srow[rr];
        h1[row * HID + nA + lo] = fmaxf(c0[r] + s * bias[nA + lo], 0.0f);
        h1[row * HID + nB + lo] = fmaxf(c1[r] + s * bias[nB + lo], 0.0f);
    }
}

// ---------------- fused GEMM2: out = (h1+agg1/max(deg,1)) @ Wout + bout ----------------
// block = 256 thr; 16-row M tile (guarded, N2 not multiple of 16); 8 waves = 8 N tiles (128 cols)
__global__ void __launch_bounds__(256)
gemm2_fused(const float* __restrict__ h1, const float* __restrict__ agg1,
            const float* __restrict__ deg1, const float* __restrict__ W,
            const float* __restrict__ bias, float* __restrict__ out) {
    __shared__ float As[16 * 260];  // stride 260 to avoid bank conflicts
    const int m0 = blockIdx.x * 16;
    const int tid = threadIdx.x;
#pragma unroll
    for (int i = 0; i < 16; ++i) {
        int idx = tid + i * 256;           // 16*256 elements
        int r = idx >> 8, c = idx & 255;
        int row = m0 + r;
        float v = 0.0f;
        if (row < N2) {
            float inv = 1.0f / fmaxf(deg1[row], 1.0f);
            v = h1[(long)row * HID + c] + agg1[(long)row * HID + c] * inv;
        }
        As[r * 260 + c] = v;
    }
    __syncthreads();

    const int lane = tid & 31, wave = tid >> 5;
    const int lo = lane & 15, hi = lane >> 4;
    const int n0 = wave * 16;
    v8f acc = {0, 0, 0, 0, 0, 0, 0, 0};
    for (int k = 0; k < HID; k += 4) {
        const int kk = k + 2 * hi;
        v2f a;
        a[0] = As[lo * 260 + kk];
        a[1] = As[lo * 260 + kk + 1];
        v2f b;
        b[0] = W[kk * DOUT + n0 + lo];
        b[1] = W[(kk + 1) * DOUT + n0 + lo];
        acc = __builtin_amdgcn_wmma_f32_16x16x4_f32(false, a, false, b, (short)0, acc, false, false);
    }
#pragma unroll
    for (int r = 0; r < 8; ++r) {
        int row = m0 + r + 8 * hi;
        if (row < N2) out[(long)row * DOUT + n0 + lo] = acc[r] + bias[n0 + lo];
    }
}

// ---------------- launcher ----------------
extern "C" void kernel_launch(void* const* d_in, const int* in_sizes, int n_in,
                              void* d_out, int out_size, void* d_ws, size_t ws_size,
                              hipStream_t stream) {
    (void)in_sizes; (void)n_in; (void)out_size; (void)ws_size;
    const int*   x  = (const int*)d_in[0];
    const float* e0 = (const float*)d_in[1];
    const float* e1 = (const float*)d_in[2];
    const float* e2 = (const float*)d_in[3];
    const float* e3 = (const float*)d_in[4];
    const float* Win[2]  = {(const float*)d_in[5], (const float*)d_in[9]};
    const float* bin[2]  = {(const float*)d_in[6], (const float*)d_in[10]};
    const float* Wout[2] = {(const float*)d_in[7], (const float*)d_in[11]};
    const float* bout[2] = {(const float*)d_in[8], (const float*)d_in[12]};
    const int* srcL0[2] = {(const int*)d_in[13], (const int*)d_in[17]};
    const int* dstL0[2] = {(const int*)d_in[14], (const int*)d_in[18]};
    const int* srcL1[2] = {(const int*)d_in[15], (const int*)d_in[19]};
    const int* dstL1[2] = {(const int*)d_in[16], (const int*)d_in[20]};
    float* out = (float*)d_out;

    // workspace layout (256B aligned)
    size_t off = 0;
    auto take = [&](size_t bytes) { size_t o = off; off += (bytes + 255) & ~(size_t)255; return o; };
    char* ws = (char*)d_ws;
    float* feat = (float*)(ws + take((size_t)N0 * DIN * 4));
    float* h1   = (float*)(ws + take((size_t)N1 * HID * 4));
    size_t zbeg = off;
    float* agg0 = (float*)(ws + take((size_t)N1 * DIN * 4));
    float* deg0 = (float*)(ws + take((size_t)N1 * 4));
    float* agg1 = (float*)(ws + take((size_t)N2 * HID * 4));
    float* deg1 = (float*)(ws + take((size_t)N2 * 4));
    float* zptr = (float*)(ws + zbeg);
    long   zcnt = (long)((off - zbeg) / 4);

    // feat once (shared by both branches): one wave/row, 8 rows per 256-thr block
    build_feat<<<N0 / 8, 256, 0, stream>>>(x, e0, e1, e2, e3, feat);

    for (int e = 0; e < 2; ++e) {
        zero_f32<<<(unsigned)((zcnt + 255) / 256), 256, 0, stream>>>(zptr, zcnt);
        scatter_edges<DIN><<<E0 / 8, 256, 0, stream>>>(feat, srcL0[e], dstL0[e], agg0, deg0, E0);
        gemm1_fused<<<N1 / 16, 256, 0, stream>>>(feat, agg0, deg0, Win[e], bin[e], h1);
        scatter_edges<HID><<<E1 / 8, 256, 0, stream>>>(h1, srcL1[e], dstL1[e], agg1, deg1, E1);
        gemm2_fused<<<(N2 + 15) / 16, 256, 0, stream>>>(h1, agg1, deg1, Wout[e], bout[e],
                                                        out + (size_t)e * N2 * DOUT);
    }
}